// BesselFilter_9947144257666
// MI455X (gfx1250) — compile-verified
//
#include <hip/hip_runtime.h>

typedef __attribute__((ext_vector_type(2))) float v2f;
typedef __attribute__((ext_vector_type(8))) float v8f;

// Problem constants (match reference)
constexpr int ROWS  = 64;
constexpr int TLEN  = 131072;
constexpr int NTAP  = 6;            // order-5 filter -> 6 coeffs

// Truncated-IIR-as-FIR parameters: max |z-pole| ~= 0.60 -> 97+ taps is
// ~1e-21 truncation error, far below fp32 ulp.
constexpr int CBLK  = 7;            // 7 Toeplitz blocks of 16 taps
constexpr int HTAPS = CBLK * 16;    // 112 taps

// Tiling
constexpr int WAVES      = 8;                       // 256-thread block = 8 wave32
constexpr int WAVE_TILES = 4;                       // 16-wide output tiles per wave
constexpr int BLOCK_SPAN = WAVES * WAVE_TILES * 16; // 512 time samples per block
constexpr int SX_COLS    = BLOCK_SPAN + HTAPS;      // 624 staged columns
constexpr int SX_STRIDE  = SX_COLS + 1;             // 625 (odd -> LDS conflict-free)

// ---------------------------------------------------------------------------
// CDNA5 async global->LDS copy helpers (ASYNCcnt-tracked, bypasses VGPRs).
// b32 flavor on purpose: keeps the odd (bank-conflict-free) LDS stride legal.
// Builtin prototype (per hipcc diagnostic): (global int*, lds int*, imm, imm).
// ---------------------------------------------------------------------------
typedef __attribute__((address_space(1))) int glb_i32_t;
typedef __attribute__((address_space(3))) int lds_i32_t;

__device__ __forceinline__ void async_copy_b32(float* lds_dst, const float* gsrc) {
#if __has_builtin(__builtin_amdgcn_global_load_async_to_lds_b32)
  __builtin_amdgcn_global_load_async_to_lds_b32(
      (glb_i32_t*)gsrc, (lds_i32_t*)lds_dst, /*offset=*/0, /*cpol=*/0);
#else
  asm volatile("global_load_async_to_lds_b32 %0, %1, off"
               :
               : "v"((unsigned)(unsigned long long)lds_dst), "v"(gsrc)
               : "memory");
#endif
}

__device__ __forceinline__ void wait_asynccnt0() {
#if __has_builtin(__builtin_amdgcn_s_wait_asynccnt)
  __builtin_amdgcn_s_wait_asynccnt(0);
#else
  asm volatile("s_wait_asynccnt 0x0" ::: "memory");
#endif
}

// ---------------------------------------------------------------------------
// Kernel 1: materialize the combined impulse response h = (b/a0) * IR(1/(a/a0))
// Serial 112-step recurrence; negligible cost, runs once.
// ---------------------------------------------------------------------------
__global__ void bessel_ir_kernel(const float* __restrict__ b,
                                 const float* __restrict__ a,
                                 float* __restrict__ h) {
  if (threadIdx.x == 0 && blockIdx.x == 0) {
    float bn[NTAP], an[NTAP];
    const float a0 = a[0];
    for (int i = 0; i < NTAP; ++i) { bn[i] = b[i] / a0; an[i] = a[i] / a0; }
    float hb[HTAPS];
    for (int t = 0; t < HTAPS; ++t) {
      float v = (t < NTAP) ? bn[t] : 0.0f;
      for (int k = 1; k < NTAP; ++k)
        if (t - k >= 0) v -= an[k] * hb[t - k];
      hb[t] = v;
      h[t]  = v;
    }
  }
}

// ---------------------------------------------------------------------------
// Kernel 2: y = conv(x, h) (causal, zero-padded) via V_WMMA_F32_16X16X4_F32.
// Output tile = 16 time steps (M) x 16 batch rows (N); K = filter taps.
// ---------------------------------------------------------------------------
__global__ __launch_bounds__(256)
void bessel_conv_wmma_kernel(const float* __restrict__ x,
                             const float* __restrict__ h,
                             float* __restrict__ y) {
  __shared__ float sx[16 * SX_STRIDE];                 // staged x (+halo)
  __shared__ float sy[WAVES][16][WAVE_TILES * 16];     // output transpose buffer
  __shared__ float sh[HTAPS];                          // impulse response

  const int tid     = threadIdx.x;
  const int lane    = tid & 31;
  const int wave    = tid >> 5;
  const int rowbase = blockIdx.y * 16;
  const int t_block = blockIdx.x * BLOCK_SPAN;

  if (tid < HTAPS) sh[tid] = h[tid];

  // Stage x[16 rows][t_block - HTAPS .. t_block + BLOCK_SPAN) into LDS.
  // Interior blocks: async global->LDS copies (ASYNCcnt), every t in range.
  // Block 0: plain path with zero fill for the causal t<0 halo.
  if (blockIdx.x > 0) {
    for (int i = tid; i < 16 * SX_COLS; i += 256) {
      const int r = i / SX_COLS;
      const int c = i - r * SX_COLS;
      const int t = t_block - HTAPS + c;
      async_copy_b32(&sx[r * SX_STRIDE + c],
                     &x[(size_t)(rowbase + r) * TLEN + t]);
    }
    wait_asynccnt0();
  } else {
    for (int i = tid; i < 16 * SX_COLS; i += 256) {
      const int r = i / SX_COLS;
      const int c = i - r * SX_COLS;
      const int t = t_block - HTAPS + c;
      float v = 0.0f;
      if (t >= 0) v = x[(size_t)(rowbase + r) * TLEN + t];
      sx[r * SX_STRIDE + c] = v;
    }
  }
  __syncthreads();

  // Constant Toeplitz A operands (held in VGPRs for the whole kernel).
  // A 16x4 fp32 layout: lane holds M = lane&15; K = 2*(lane>>4) + elem.
  const int m  = lane & 15;          // M for A, N for B/D (all use lane&15)
  const int kb = (lane >> 4) << 1;   // K base for this half-wave
  v2f aop[CBLK][4];
  #pragma unroll
  for (int c = 0; c < CBLK; ++c) {
    #pragma unroll
    for (int j = 0; j < 4; ++j) {
      const int i0 = 16 * c + m - 4 * j - kb;   // tap index for elem 0
      const int i1 = i0 - 1;                    // tap index for elem 1
      v2f v;
      v.x = (i0 >= 0 && i0 < HTAPS) ? sh[i0] : 0.0f;
      v.y = (i1 >= 0 && i1 < HTAPS) ? sh[i1] : 0.0f;
      aop[c][j] = v;
    }
  }

  // Each wave computes WAVE_TILES consecutive output tiles.
  const int wave_t0 = wave * (WAVE_TILES * 16);
  const float* srow = &sx[m * SX_STRIDE];   // B operand: N = row = lane&15
  #pragma unroll
  for (int tile = 0; tile < WAVE_TILES; ++tile) {
    const int t0l = wave_t0 + tile * 16;    // local time of first output
    v8f acc = {};
    #pragma unroll
    for (int c = 0; c < CBLK; ++c) {
      const int u = t0l - 16 * c + HTAPS;   // LDS column of J=0 for block c
      #pragma unroll
      for (int j = 0; j < 4; ++j) {
        const int col = u + 4 * j + kb;     // B 4x16: VGPRv -> K = kb + v
        v2f bop;
        bop.x = srow[col];
        bop.y = srow[col + 1];
        // D(16x16,f32) += A(16x4,f32) x B(4x16,f32)
        acc = __builtin_amdgcn_wmma_f32_16x16x4_f32(
            /*neg_a=*/false, aop[c][j],
            /*neg_b=*/false, bop,
            /*c_mod=*/(short)0, acc,
            /*reuse_a=*/false, /*reuse_b=*/false);
      }
    }
    // D layout: VGPR r -> (M = r + 8*(lane>>4), N = lane&15)
    const int mhalf = (lane >> 4) << 3;
    #pragma unroll
    for (int r = 0; r < 8; ++r)
      sy[wave][m][tile * 16 + mhalf + r] = acc[r];
  }
  __builtin_amdgcn_wave_barrier();

  // Coalesced row-major store: 16 rows x 64 floats per wave, float4 chunks.
  const int t_wave = t_block + wave_t0;
  for (int i = lane; i < 16 * 16; i += 32) {
    const int r  = i >> 4;   // row 0..15
    const int c4 = i & 15;   // float4 index 0..15
    const float4 v = *(const float4*)&sy[wave][r][c4 * 4];
    *(float4*)&y[(size_t)(rowbase + r) * TLEN + t_wave + c4 * 4] = v;
  }
}

// ---------------------------------------------------------------------------
// Harness entry point.
// Inputs: d_in[0]=x (64*131072 f32), d_in[1]=b (6 f32), d_in[2]=a (6 f32).
// Output: y (64*131072 f32). The reference's xmax normalization cancels
// exactly by linearity (xmax * lfilter(x/xmax) == lfilter(x)), so no global
// max reduction is needed.
// ---------------------------------------------------------------------------
extern "C" void kernel_launch(void* const* d_in, const int* in_sizes, int n_in,
                              void* d_out, int out_size, void* d_ws, size_t ws_size,
                              hipStream_t stream) {
  (void)in_sizes; (void)n_in; (void)out_size; (void)ws_size;
  const float* x = (const float*)d_in[0];
  const float* b = (const float*)d_in[1];
  const float* a = (const float*)d_in[2];
  float* y = (float*)d_out;
  float* h = (float*)d_ws;   // 112 floats of scratch

  bessel_ir_kernel<<<1, 32, 0, stream>>>(b, a, h);

  dim3 grid(TLEN / BLOCK_SPAN, ROWS / 16);   // (256, 4)
  bessel_conv_wmma_kernel<<<grid, 256, 0, stream>>>(x, h, y);
}